// MetaPathGNN_1675037245544
// MI455X (gfx1250) — compile-verified
//
#include <hip/hip_runtime.h>
#include <math.h>

typedef __attribute__((ext_vector_type(2))) float v2f;
typedef __attribute__((ext_vector_type(8))) float v8f;

#define NNODES 50000
#define NEDGES 800000

// ---------------------------------------------------------------------------
// WMMA f32 GEMM:  C = act( A1[M,K1]@B1[K1,N] (+ A2[M,K2]@B2[K2,N]) + bias )
// Row-major. Each wave computes a 16x64 tile with V_WMMA_F32_16X16X4_F32
// (4 accumulators). The 64-col weight tile is staged into LDS pre-swizzled
// into per-lane fragment order (float2 @ index k4*128 + t*32 + lane), so the
// hot loop is: 1x global_load_b64 (A frag, pointer-bumped) + 4x conflict-free
// ds_load_b64 (B frags) + 4x v_wmma. Dynamic LDS = K*64 floats (<=64KB).
// Invalid trailing waves clamp their A row and skip stores, so every wave
// reaches the barriers and EXEC is all-ones at each WMMA (ISA requirement).
// ---------------------------------------------------------------------------
__global__ __launch_bounds__(128) void gemm_wmma_kernel(
    const float* __restrict__ A1, const float* __restrict__ B1, int K1,
    const float* __restrict__ A2, const float* __restrict__ B2, int K2,
    const float* __restrict__ bias, float* __restrict__ Cout,
    int M, int Ncols, int do_relu)
{
  extern __shared__ float smem[];
  const int wave = threadIdx.x >> 5;
  const int lane = threadIdx.x & 31;
  const int row0 = blockIdx.x * 64 + wave * 16;   // wave-uniform
  const int col0 = blockIdx.y * 64;
  const bool valid = (row0 < M);                  // M % 16 == 0 here

  const int mloc = lane & 15;                     // A: M index / B: N index
  const int kb   = (lane >> 4) << 1;              // 0 or 2 (K sub-offset)
  const int rowa = valid ? (row0 + mloc) : mloc;  // clamp keeps loads in-bounds
  v8f acc[4] = {};

  // ---------------- pass 1: A1 @ B1 ----------------
  {
    const int nf2 = K1 << 5;                      // K1*32 float2 fragments
    for (int f = threadIdx.x; f < nf2; f += 128) {
      const int l  = f & 31;
      const int t  = (f >> 5) & 3;
      const int k4 = f >> 7;
      const int kr = (k4 << 2) + ((l >> 4) << 1);
      const size_t src = (size_t)kr * Ncols + col0 + t * 16 + (l & 15);
      smem[f * 2 + 0] = B1[src];
      smem[f * 2 + 1] = B1[src + Ncols];
    }
    __syncthreads();
    const float* __restrict__ ap = A1 + (size_t)rowa * K1 + kb;
    const float2* __restrict__ bf = (const float2*)smem;
    int fb = lane;
    for (int k4 = 0; k4 < (K1 >> 2); ++k4) {
      v2f a; a.x = ap[0]; a.y = ap[1]; ap += 4;
#pragma unroll
      for (int t = 0; t < 4; ++t) {
        const float2 bv = bf[fb + t * 32];
        v2f b; b.x = bv.x; b.y = bv.y;
        acc[t] = __builtin_amdgcn_wmma_f32_16x16x4_f32(
            false, a, false, b, (short)0, acc[t], false, false);
      }
      fb += 128;
    }
  }

  // ---------------- pass 2 (optional): + A2 @ B2 ----------------
  if (A2) {  // uniform branch
    __syncthreads();
    const int nf2 = K2 << 5;
    for (int f = threadIdx.x; f < nf2; f += 128) {
      const int l  = f & 31;
      const int t  = (f >> 5) & 3;
      const int k4 = f >> 7;
      const int kr = (k4 << 2) + ((l >> 4) << 1);
      const size_t src = (size_t)kr * Ncols + col0 + t * 16 + (l & 15);
      smem[f * 2 + 0] = B2[src];
      smem[f * 2 + 1] = B2[src + Ncols];
    }
    __syncthreads();
    const float* __restrict__ ap = A2 + (size_t)rowa * K2 + kb;
    const float2* __restrict__ bf = (const float2*)smem;
    int fb = lane;
    for (int k4 = 0; k4 < (K2 >> 2); ++k4) {
      v2f a; a.x = ap[0]; a.y = ap[1]; ap += 4;
#pragma unroll
      for (int t = 0; t < 4; ++t) {
        const float2 bv = bf[fb + t * 32];
        v2f b; b.x = bv.x; b.y = bv.y;
        acc[t] = __builtin_amdgcn_wmma_f32_16x16x4_f32(
            false, a, false, b, (short)0, acc[t], false, false);
      }
      fb += 128;
    }
  }

  if (!valid) return;
  // C/D layout: VGPR r -> rows (r, r+8); lane halves split M by 8
  const int rbase = row0 + ((lane >> 4) << 3);
#pragma unroll
  for (int t = 0; t < 4; ++t) {
    const int col = col0 + t * 16 + mloc;
    const float bv = bias ? bias[col] : 0.0f;
#pragma unroll
    for (int r = 0; r < 8; ++r) {
      float v = acc[t][r] + bv;
      if (do_relu) v = fmaxf(v, 0.0f);
      Cout[(size_t)(rbase + r) * Ncols + col] = v;
    }
  }
}

// ---------------------------------------------------------------------------
// Edge-filtered gather + atomic scatter-add:
//   aggr[dst] += feat[src]   for edges with edge_type == rel
// 2^tpe_shift threads per edge, float4 per thread, native f32 atomics.
// ---------------------------------------------------------------------------
__global__ void edge_agg_kernel(const float* __restrict__ feat,
                                float* __restrict__ aggr,
                                const int* __restrict__ edge_index,
                                const int* __restrict__ edge_type,
                                int rel, int tpe_shift, int dimf)
{
  const long long gtid = (long long)blockIdx.x * blockDim.x + threadIdx.x;
  const long long edge = gtid >> tpe_shift;
  if (edge >= NEDGES) return;
  if (edge_type[edge] != rel) return;
  const int c4  = (int)(gtid & ((1 << tpe_shift) - 1)) << 2;
  const int dst = edge_index[edge];            // row 0: aggregation target
  const int src = edge_index[NEDGES + edge];   // row 1: message source (x_j)
  const float4 v = *reinterpret_cast<const float4*>(feat + (size_t)src * dimf + c4);
  float* out = aggr + (size_t)dst * dimf + c4;
  unsafeAtomicAdd(out + 0, v.x);
  unsafeAtomicAdd(out + 1, v.y);
  unsafeAtomicAdd(out + 2, v.z);
  unsafeAtomicAdd(out + 3, v.w);
}

__global__ void zero_kernel(float* __restrict__ p, size_t n) {
  size_t i = (size_t)blockIdx.x * blockDim.x + threadIdx.x;
  const size_t stride = (size_t)gridDim.x * blockDim.x;
  for (; i < n; i += stride) p[i] = 0.0f;
}

__global__ void add2_kernel(float* __restrict__ d, const float* __restrict__ a,
                            const float* __restrict__ b, int n) {
  int i = blockIdx.x * blockDim.x + threadIdx.x;
  if (i < n) d[i] = a[i] + b[i];
}

__global__ void add3_kernel(float* __restrict__ d, const float* __restrict__ a,
                            const float* __restrict__ b, const float* __restrict__ c, int n) {
  int i = blockIdx.x * blockDim.x + threadIdx.x;
  if (i < n) d[i] = a[i] + b[i] + c[i];
}

// ---------------------------------------------------------------------------
// logits = z[N,128] @ fc2_w[128,40] + fc2_b ; out = log_softmax(logits)
// One wave32 per row; lane handles cols {lane, lane+32(lane<8)}.
// ---------------------------------------------------------------------------
__global__ __launch_bounds__(256) void fc2_logsoftmax_kernel(
    const float* __restrict__ z, const float* __restrict__ w,
    const float* __restrict__ b, float* __restrict__ out, int nrows)
{
  const int row  = blockIdx.x * (blockDim.x >> 5) + (threadIdx.x >> 5);
  const int lane = threadIdx.x & 31;
  if (row >= nrows) return;
  const float* zr = z + (size_t)row * 128;
  const int c0 = lane, c1 = 32 + lane;          // c1 valid for lane < 8
  float acc0 = 0.0f, acc1 = 0.0f;
  for (int k = 0; k < 128; ++k) {
    const float zv = zr[k];
    acc0 += zv * w[k * 40 + c0];
    if (lane < 8) acc1 += zv * w[k * 40 + c1];
  }
  acc0 += b[c0];
  if (lane < 8) acc1 += b[c1];

  float mx = acc0;
  if (lane < 8) mx = fmaxf(mx, acc1);
  for (int off = 16; off; off >>= 1) mx = fmaxf(mx, __shfl_xor(mx, off, 32));
  float s = expf(acc0 - mx) + ((lane < 8) ? expf(acc1 - mx) : 0.0f);
  for (int off = 16; off; off >>= 1) s += __shfl_xor(s, off, 32);
  const float lse = mx + logf(s);

  out[(size_t)row * 40 + c0] = acc0 - lse;
  if (lane < 8) out[(size_t)row * 40 + c1] = acc1 - lse;
}

// ---------------------------------------------------------------------------
extern "C" void kernel_launch(void* const* d_in, const int* in_sizes, int n_in,
                              void* d_out, int out_size, void* d_ws, size_t ws_size,
                              hipStream_t stream) {
  (void)in_sizes; (void)n_in; (void)out_size; (void)ws_size;
  const float* x     = (const float*)d_in[0];
  const int*   eidx  = (const int*)d_in[1];
  const int*   etype = (const int*)d_in[2];
  const float* mw1 = (const float*)d_in[3];  const float* mb1 = (const float*)d_in[4];
  const float* mw2 = (const float*)d_in[5];  const float* mb2 = (const float*)d_in[6];
  const float* mw3 = (const float*)d_in[7];  const float* mb3 = (const float*)d_in[8];
  const float* w0_0 = (const float*)d_in[9];  const float* b0_0 = (const float*)d_in[10];
  const float* wl_0 = (const float*)d_in[11]; const float* bl_0 = (const float*)d_in[12];
  const float* w1_0 = (const float*)d_in[13]; const float* b1_0 = (const float*)d_in[14];
  const float* w0_1 = (const float*)d_in[15]; const float* b0_1 = (const float*)d_in[16];
  const float* wl_1 = (const float*)d_in[17]; const float* bl_1 = (const float*)d_in[18];
  const float* w1_1 = (const float*)d_in[19]; const float* b1_1 = (const float*)d_in[20];
  const float* fc1w = (const float*)d_in[21]; const float* fc1b = (const float*)d_in[22];
  const float* fc2w = (const float*)d_in[23]; const float* fc2b = (const float*)d_in[24];
  float* out = (float*)d_out;

  // Workspace layout (floats); peak ~154 MB
  const size_t N = NNODES;
  float* ws    = (float*)d_ws;
  float* bufA  = ws;                   // N*256 : h, later emb2
  float* bufB  = bufA + N * 256;       // N*256 : aggr0, later z
  float* bufC  = bufB + N * 256;       // N*128 : h1, later emb
  float* bufD  = bufC + N * 128;       // N*128 : h2, later aggr1
  float* w10_0 = bufD + N * 128;       // 256*128 : w1_0 + w0_0
  float* w10_1 = w10_0 + 256 * 128;    // 128*128 : w1_1 + w0_1
  float* fc1c  = w10_1 + 128 * 128;    // 128*128 : fc1_w[:128] + fc1_w[128:]
  float* bsum0 = fc1c + 128 * 128;     // 128
  float* bsum1 = bsum0 + 128;          // 128

  // --- fold weights/biases (metapaths identical -> fc1 halves fold too) ---
  add2_kernel<<<(32768 + 255) / 256, 256, 0, stream>>>(w10_0, w1_0, w0_0, 256 * 128);
  add2_kernel<<<(16384 + 255) / 256, 256, 0, stream>>>(w10_1, w1_1, w0_1, 128 * 128);
  add2_kernel<<<(16384 + 255) / 256, 256, 0, stream>>>(fc1c, fc1w, fc1w + 128 * 128, 128 * 128);
  add3_kernel<<<1, 128, 0, stream>>>(bsum0, bl_0, b1_0, b0_0, 128);
  add3_kernel<<<1, 128, 0, stream>>>(bsum1, bl_1, b1_1, b0_1, 128);

  const dim3 g2((NNODES + 63) / 64, 2);  // Ncols = 128
  const dim3 g4((NNODES + 63) / 64, 4);  // Ncols = 256
  const size_t sh128 = 128 * 64 * sizeof(float);  // 32 KB (K=128)
  const size_t sh256 = 256 * 64 * sizeof(float);  // 64 KB (K=256)

  // --- MLP: h = (relu(relu(x@W1+b1)@W2+b2))@W3+b3  -> bufA [N,256] ---
  gemm_wmma_kernel<<<g2, 128, sh128, stream>>>(x,    mw1, 128, nullptr, nullptr, 0, mb1, bufC, NNODES, 128, 1);
  gemm_wmma_kernel<<<g2, 128, sh128, stream>>>(bufC, mw2, 128, nullptr, nullptr, 0, mb2, bufD, NNODES, 128, 1);
  gemm_wmma_kernel<<<g4, 128, sh128, stream>>>(bufD, mw3, 128, nullptr, nullptr, 0, mb3, bufA, NNODES, 256, 0);

  // --- layer 0 (rel=2, D=256): aggr0 = scatter(h) ; emb = relu(aggr0@wl_0 + h@w10_0 + bsum0)
  zero_kernel<<<2048, 256, 0, stream>>>(bufB, N * 256);
  edge_agg_kernel<<<(int)(((long long)NEDGES * 64 + 255) / 256), 256, 0, stream>>>(
      bufA, bufB, eidx, etype, 2, 6, 256);
  gemm_wmma_kernel<<<g2, 128, sh256, stream>>>(bufB, wl_0, 256, bufA, w10_0, 256, bsum0, bufC, NNODES, 128, 1);

  // --- layer 1 (rel=3, D=128): aggr1 = scatter(emb) ; emb2 = relu(aggr1@wl_1 + emb@w10_1 + bsum1)
  zero_kernel<<<2048, 256, 0, stream>>>(bufD, N * 128);
  edge_agg_kernel<<<(int)(((long long)NEDGES * 32 + 255) / 256), 256, 0, stream>>>(
      bufC, bufD, eidx, etype, 3, 5, 128);
  gemm_wmma_kernel<<<g2, 128, sh128, stream>>>(bufD, wl_1, 128, bufC, w10_1, 128, bsum1, bufA, NNODES, 128, 1);

  // --- fc1 on folded weight: z = relu(emb2 @ fc1c + fc1_b) ---
  gemm_wmma_kernel<<<g2, 128, sh128, stream>>>(bufA, fc1c, 128, nullptr, nullptr, 0, fc1b, bufB, NNODES, 128, 1);

  // --- fc2 + log_softmax -> d_out [N,40] ---
  fc2_logsoftmax_kernel<<<(NNODES + 7) / 8, 256, 0, stream>>>(bufB, fc2w, fc2b, out, NNODES);
}